// PowerApproximationLayer_45741401702555
// MI455X (gfx1250) — compile-verified
//
#include <hip/hip_runtime.h>

// ---------------------------------------------------------------------------
// PowerApproximationLayer on MI455X (gfx1250)
//   199 x { w_new = scatter_add(ew * w[src] -> dst); out += w_new }
// L2-resident streaming SpMV; fp32; async global->LDS staging of edge tiles.
// (pa_scatter placed first so the disasm snippet shows the async path.)
// ---------------------------------------------------------------------------

#define THREADS 256
#define TILE    1024            // edges per tile (THREADS * VEC)
#define VEC     4               // b128 = 4 dwords per lane
#define MAX_GRID 2048
#define NUM_ITERS 200           // matches reference NUM_ITERATIONS

#if __has_builtin(__builtin_amdgcn_global_load_async_to_lds_b128)
#define HAVE_ASYNC_BUILTIN 1
#else
#define HAVE_ASYNC_BUILTIN 0
#endif

// Pointee type the builtin expects (per hipcc diagnostic): int vector_size(16)
typedef int vint4 __attribute__((vector_size(16)));
typedef __attribute__((address_space(1))) vint4* gptr_b128;
typedef __attribute__((address_space(3))) vint4* lptr_b128;

// ---- CDNA5 async global->LDS copy (16B per active lane, per-lane LDS addr) -
__device__ __forceinline__ void async_copy_b128(const void* gsrc, void* ldst) {
#if HAVE_ASYNC_BUILTIN
    __builtin_amdgcn_global_load_async_to_lds_b128(
        (gptr_b128)(unsigned long long)gsrc,
        (lptr_b128)(unsigned int)(unsigned long long)ldst,
        0, 0);
#else
    // Fallback: raw CDNA5 mnemonic, GV addressing (64-bit VGPR address pair).
    unsigned int l = (unsigned int)(unsigned long long)ldst;  // LDS byte offset
    asm volatile("global_load_async_to_lds_b128 %0, %1, off"
                 :: "v"(l), "v"(gsrc)
                 : "memory");
#endif
}

__device__ __forceinline__ void wait_async_le3() {
#if __has_builtin(__builtin_amdgcn_s_wait_asynccnt)
    __builtin_amdgcn_s_wait_asynccnt(3);
#else
    asm volatile("s_wait_asynccnt 0x3" ::: "memory");
#endif
    asm volatile("" ::: "memory");   // keep LDS reads below the wait
}

__device__ __forceinline__ void wait_async_le0() {
#if __has_builtin(__builtin_amdgcn_s_wait_asynccnt)
    __builtin_amdgcn_s_wait_asynccnt(0);
#else
    asm volatile("s_wait_asynccnt 0x0" ::: "memory");
#endif
    asm volatile("" ::: "memory");
}

// ---- one SpMV step: wn[dst] += ew * w[src] --------------------------------
// Persistent blocks, grid-stride over 1024-edge tiles. Each wave async-stages
// its own lanes' slice of (dst, src, ew) into LDS and consumes only what it
// staged -> per-wave s_wait_asynccnt is the only sync needed (no barriers).
// Double-buffered: while tile t is processed, tile t+grid is in flight.
__global__ __launch_bounds__(THREADS)
void pa_scatter(const int*   __restrict__ edst,
                const int*   __restrict__ esrc,
                const float* __restrict__ ewt,
                const float* __restrict__ w,
                float*       __restrict__ wn,
                int nEdges, int nFullTiles) {
    __shared__ int   sd[2][TILE];
    __shared__ int   ss[2][TILE];
    __shared__ float sw[2][TILE];

    const int tid    = threadIdx.x;
    const int off    = tid * VEC;
    const int stride = gridDim.x;

    int tile = blockIdx.x;
    int buf  = 0;
    if (tile < nFullTiles) {
        long e0 = (long)tile * TILE + off;
        async_copy_b128(edst + e0, &sd[0][off]);
        async_copy_b128(esrc + e0, &ss[0][off]);
        async_copy_b128(ewt  + e0, &sw[0][off]);
    }

    for (; tile < nFullTiles; tile += stride) {
        int nextTile = tile + stride;
        if (nextTile < nFullTiles) {
            long e1 = (long)nextTile * TILE + off;
            int nb = buf ^ 1;
            async_copy_b128(edst + e1, &sd[nb][off]);
            async_copy_b128(esrc + e1, &ss[nb][off]);
            async_copy_b128(ewt  + e1, &sw[nb][off]);
            wait_async_le3();   // 3 newest (next tile) may stay in flight
        } else {
            wait_async_le0();   // last tile: drain everything
        }

#pragma unroll
        for (int k = 0; k < VEC; ++k) {
            int   d = sd[buf][off + k];
            int   s = ss[buf][off + k];
            float v = sw[buf][off + k] * w[s];     // gather hits L2 (400KB vec)
            atomicAdd(&wn[d], v);                  // global_atomic_add_f32, no-rtn
        }
        buf ^= 1;
    }

    // tail edges if E % TILE != 0 (not hit for E = 6.4M, kept for safety)
    long tail = (long)nFullTiles * TILE;
    for (long e = tail + (long)blockIdx.x * THREADS + tid; e < nEdges;
         e += (long)stride * THREADS) {
        atomicAdd(&wn[edst[e]], ewt[e] * w[esrc[e]]);
    }
}

// ---- iteration 0: w = mask ? x : 0 ; out = w ; w_new = 0 -------------------
__global__ __launch_bounds__(THREADS)
void pa_init(const float* __restrict__ x, const int* __restrict__ mask,
             float* __restrict__ w, float* __restrict__ wn,
             float* __restrict__ out, int n) {
    int i = blockIdx.x * blockDim.x + threadIdx.x;
    if (i < n) {
        float v = mask[i] ? x[i] : 0.0f;
        w[i]   = v;
        out[i] = v;
        wn[i]  = 0.0f;
    }
}

// ---- out += w_new ; zero the buffer that becomes next round's w_new --------
__global__ __launch_bounds__(THREADS)
void pa_accum_zero(const float* __restrict__ wn, float* __restrict__ out,
                   float* __restrict__ wz, int n) {
    int i = blockIdx.x * blockDim.x + threadIdx.x;
    if (i < n) {
        out[i] += wn[i];
        wz[i]   = 0.0f;
    }
}

// ---------------------------------------------------------------------------
extern "C" void kernel_launch(void* const* d_in, const int* in_sizes, int n_in,
                              void* d_out, int out_size, void* d_ws, size_t ws_size,
                              hipStream_t stream) {
    const float* x    = (const float*)d_in[0];
    const int*   eidx = (const int*)  d_in[1];   // [2, E] row-major
    const float* ew   = (const float*)d_in[2];
    const int*   mask = (const int*)  d_in[3];

    const int n = in_sizes[0];
    const int E = in_sizes[1] / 2;
    const int* edst = eidx;        // edge_index[0] = dst
    const int* esrc = eidx + E;    // edge_index[1] = src

    float* bufA = (float*)d_ws;    // ping-pong w / w_new (2 * N floats)
    float* bufB = bufA + n;
    float* out  = (float*)d_out;

    const int nBlocks = (n + THREADS - 1) / THREADS;
    pa_init<<<nBlocks, THREADS, 0, stream>>>(x, mask, bufA, bufB, out, n);

    const int fullTiles = E / TILE;
    int grid = fullTiles < MAX_GRID ? fullTiles : MAX_GRID;
    if (grid < 1) grid = 1;

    float* w  = bufA;
    float* wn = bufB;
    for (int t = 1; t < NUM_ITERS; ++t) {
        pa_scatter<<<grid, THREADS, 0, stream>>>(edst, esrc, ew, w, wn, E, fullTiles);
        pa_accum_zero<<<nBlocks, THREADS, 0, stream>>>(wn, out, w, n);
        float* tmp = w; w = wn; wn = tmp;   // old w is now zeroed -> next w_new
    }
}